// Conv2DSum_1700807049807
// MI455X (gfx1250) — compile-verified
//
#include <hip/hip_runtime.h>

// SPN log-conv 1x1:  out[m,s] = max_c x[m,c] + log( sum_c exp(x[m,c]-max) * w[c,s] )
// w = softmax(acc over c).  M = 524288, Cin = S = 32.  Memory-bound: 128 MB @ 23.3 TB/s
// => ~5.5 us floor.  Contraction via V_WMMA_F32_16X16X4_F32 (full f32 precision).
// B fragments pinned in VGPRs (asm barrier stops LDS rematerialization); rolled
// 4-tile loop keeps the dynamic instruction stream under the per-SIMD cycle budget.

typedef float v2f __attribute__((ext_vector_type(2)));
typedef float v8f __attribute__((ext_vector_type(8)));

#define CIN   32
#define NSUM  32
#define WAVES 8          // waves per block
#define TPW   4          // row-tiles per wave

#define LOG2E 1.4426950408889634f
#define LN2   0.6931471805599453f

__global__ __launch_bounds__(256) void spn_logconv1x1_kernel(
    const float* __restrict__ x,     // [M, CIN]
    const float* __restrict__ acc,   // [CIN, NSUM]
    float* __restrict__ out,         // [M, NSUM]
    int Mrows)
{
    __shared__ float s_w[CIN * NSUM];   // normalized weights w[c][s]
    __shared__ float s_inv[NSUM];

    const int tid = threadIdx.x;

    // ---- weights: w[c][s] = exp(acc[c][s]) / sum_c exp(acc[c][s]) ----
    for (int i = tid; i < CIN * NSUM; i += 256)
        s_w[i] = __builtin_amdgcn_exp2f(acc[i] * LOG2E);
    __syncthreads();
    if (tid < NSUM) {
        float sum = 0.0f;
        #pragma unroll
        for (int c = 0; c < CIN; ++c) sum += s_w[c * NSUM + tid];
        s_inv[tid] = 1.0f / sum;
    }
    __syncthreads();
    for (int i = tid; i < CIN * NSUM; i += 256)
        s_w[i] *= s_inv[i & (NSUM - 1)];
    __syncthreads();

    const int lane   = tid & 31;
    const int wave   = tid >> 5;
    const int laneLo = lane & 15;
    const int hi     = lane >> 4;     // 0 for lanes 0-15, 1 for lanes 16-31
    const int cOff   = hi * 2;        // A/B frag: VGPR j holds K = j + 2*hi

    // ---- B fragments (4x16 f32 per k-step): read once, pinned into VGPRs ----
    v2f bfrag[2][8];
    #pragma unroll
    for (int nt = 0; nt < 2; ++nt) {
        #pragma unroll
        for (int kk = 0; kk < 8; ++kk) {
            const int c = 4 * kk + cOff;
            const int s = nt * 16 + laneLo;
            float b0 = s_w[c * NSUM + s];
            float b1 = s_w[(c + 1) * NSUM + s];
            asm volatile("" : "+v"(b0), "+v"(b1));   // forbid LDS rematerialization
            bfrag[nt][kk] = (v2f){ b0, b1 };
        }
    }

    // ---- each wave streams TPW 16-row tiles (block covers contiguous windows) ----
    int rowBase = (blockIdx.x * (WAVES * TPW) + wave) * 16;
    const float* xr = x + (size_t)(rowBase + laneLo) * CIN + cOff;
    float* op = out + (size_t)(rowBase + hi * 8) * NSUM + laneLo;
    const size_t xStep = (size_t)(WAVES * 16) * CIN;    // rows advance by 128/iter
    const size_t oStep = (size_t)(WAVES * 16) * NSUM;

    #pragma unroll 1
    for (int t = 0; t < TPW; ++t, rowBase += WAVES * 16, xr += xStep, op += oStep) {
        if (rowBase >= Mrows) break;            // wave-uniform; EXEC stays all-ones

        // A fragment loads: lane pair (L, L+16) covers row L (16 channels each)
        v2f a[8];
        #pragma unroll
        for (int kk = 0; kk < 8; ++kk)
            a[kk] = __builtin_nontemporal_load((const v2f*)(xr + 4 * kk));

        // row max: local half-row max, then merge with partner lane (lane^16)
        float m = fmaxf(a[0].x, a[0].y);
        #pragma unroll
        for (int kk = 1; kk < 8; ++kk)
            m = fmaxf(m, fmaxf(a[kk].x, a[kk].y));
        m = fmaxf(m, __shfl_xor(m, 16, 32));

        // p = exp2((x - m) * log2e)
        #pragma unroll
        for (int kk = 0; kk < 8; ++kk) {
            a[kk].x = __builtin_amdgcn_exp2f((a[kk].x - m) * LOG2E);
            a[kk].y = __builtin_amdgcn_exp2f((a[kk].y - m) * LOG2E);
        }

        // broadcast row maxes needed by the D-matrix layout rows (r + 8*hi)
        float mrow[8];
        #pragma unroll
        for (int r = 0; r < 8; ++r)
            mrow[r] = __shfl(m, r + hi * 8, 32);

        // ---- GEMM via WMMA f32 16x16x4, then log-epilogue ----
        #pragma unroll
        for (int nt = 0; nt < 2; ++nt) {
            v8f c8 = (v8f){0.f, 0.f, 0.f, 0.f, 0.f, 0.f, 0.f, 0.f};
            #pragma unroll
            for (int kk = 0; kk < 8; ++kk) {
                c8 = __builtin_amdgcn_wmma_f32_16x16x4_f32(
                    false, a[kk], false, bfrag[nt][kk], (short)0, c8, false, false);
            }
            // D layout: vgpr r -> (row = r + 8*hi, col = lane&15)
            #pragma unroll
            for (int r = 0; r < 8; ++r) {
                float v = fmaf(__builtin_amdgcn_logf(c8[r]), LN2, mrow[r]);
                __builtin_nontemporal_store(v, op + (size_t)r * NSUM + nt * 16);
            }
        }
    }
}

extern "C" void kernel_launch(void* const* d_in, const int* in_sizes, int n_in,
                              void* d_out, int out_size, void* d_ws, size_t ws_size,
                              hipStream_t stream) {
    const float* x   = (const float*)d_in[0];   // [32,128,128,32] f32
    const float* acc = (const float*)d_in[1];   // [1,1,32,32] f32
    float* out = (float*)d_out;                 // [32,128,128,32] f32

    const int M = in_sizes[0] / CIN;            // 524288 rows
    const int rowsPerBlock = WAVES * TPW * 16;  // 512
    const int blocks = (M + rowsPerBlock - 1) / rowsPerBlock;   // 1024
    spn_logconv1x1_kernel<<<blocks, 256, 0, stream>>>(x, acc, out, M);
}